// CapsuleLayer_32384053411977
// MI455X (gfx1250) — compile-verified
//
#include <hip/hip_runtime.h>

// CDNA5 / gfx1250 capsule-layer kernel.
//
// Reference collapses to 10 independent f32 GEMMs  S_n = (1/R) * X @ W_n
//   X : [B=256, K=9216]  (K = r*CIN + i)
//   W_n : [K=9216, COUT=16]
// followed by squash over the 16-wide output rows.
//
// Matrix core: V_WMMA_F32_16X16X4_F32 (full f32 precision — AI ~49 FLOP/B sits
// at the f32-matrix machine balance, so lower precision buys nothing here).

typedef __attribute__((ext_vector_type(2))) float v2f;
typedef __attribute__((ext_vector_type(8))) float v8f;

#define NCAPS   10
#define RNODES  1152
#define CIN_    8
#define COUT_   16
#define BATCH   256
#define KDIM    (RNODES * CIN_)    // 9216
#define WAVES   16                 // 512 threads / wave32
#define KCHUNK  (KDIM / WAVES)     // 576 k-values per wave

__global__ __launch_bounds__(512, 1)
void caps_wmma_kernel(const float* __restrict__ X,
                      const float* __restrict__ W,
                      float* __restrict__ out)
{
    __shared__ float red[WAVES * 256];   // 16 KB: per-wave C fragments (m,o) major

    const int tid  = threadIdx.x;
    const int wave = tid >> 5;
    const int lane = tid & 31;
    const int half = lane >> 4;          // 0: lanes 0-15, 1: lanes 16-31
    const int l16  = lane & 15;

    const int n     = blockIdx.x / (BATCH / 16);   // capsule index 0..9
    const int btile = blockIdx.x % (BATCH / 16);   // batch tile 0..15

    const int k0 = wave * KCHUNK;

    // Induction pointers (all loop loads become immediate-offset forms).
    // A-fragment row (ISA 16x4 f32 layout): lanes 0-15 -> M=lane, K = {k,k+1};
    // lanes 16-31 -> M=lane-16, K = {k+2,k+3}  => one aligned 8B load per lane.
    const float* xp = X + (size_t)(btile * 16 + l16) * KDIM + k0 + half * 2;
    // B-fragment: VGPR0 = row K / K+2 (by half), VGPR1 = row K+1 / K+3, col = l16.
    const float* wp = W + (size_t)n * KDIM * COUT_
                        + (size_t)(k0 + half * 2) * COUT_ + l16;

    v8f acc0 = {};
    v8f acc1 = {};

    // 72 iterations; 8 k-values (2 WMMA, two accumulator chains) per iteration.
    // unroll 8 => 16 WMMA per body, max immediate offset 4 KB (fits imm24).
    #pragma unroll 8
    for (int it = 0; it < KCHUNK / 8; ++it) {
        if ((it & 7) == 0) {
            // one global_prefetch_b8 per unrolled body (guard folds at compile
            // time under unroll-8): pull next 4 KB W group / next X lines toward
            // the WGP out of the (L2-resident) hierarchy.
            __builtin_prefetch(wp + 8 * 8 * COUT_, 0, 1);
            __builtin_prefetch(xp + 8 * 8, 0, 1);
        }
        // ---- step at k ----
        v2f a0 = *(const v2f*)(xp);
        v2f b0;
        b0.x = wp[0];
        b0.y = wp[COUT_];
        acc0 = __builtin_amdgcn_wmma_f32_16x16x4_f32(false, a0, false, b0,
                                                     (short)0, acc0, false, false);
        // ---- step at k+4 ----
        v2f a1 = *(const v2f*)(xp + 4);
        v2f b1;
        b1.x = wp[4 * COUT_];
        b1.y = wp[5 * COUT_];
        acc1 = __builtin_amdgcn_wmma_f32_16x16x4_f32(false, a1, false, b1,
                                                     (short)0, acc1, false, false);
        xp += 8;
        wp += 8 * COUT_;
    }

    v8f acc = acc0 + acc1;

    // Spill C fragment to LDS in (m,o)-major layout.
    // C layout: VGPR r, lanes 0-15 -> M=r, N=lane; lanes 16-31 -> M=r+8, N=lane-16.
    #pragma unroll
    for (int r = 0; r < 8; ++r) {
        red[wave * 256 + (r + 8 * half) * 16 + l16] = acc[r];
    }
    __syncthreads();

    // Cross-wave reduction + squash. 256 (m,o) items; both 256-thread halves
    // compute identically (uniform barriers), low half writes.
    const int t = tid & 255;            // t = m*16 + o
    float s = 0.0f;
    #pragma unroll
    for (int w = 0; w < WAVES; ++w) s += red[w * 256 + t];
    s *= (1.0f / (float)RNODES);        // uniform routing coefficient c = 1/R

    __syncthreads();
    if (tid < 256) red[t] = s * s;
    __syncthreads();

    const int m = t >> 4;
    float nsq = 0.0f;
    #pragma unroll
    for (int j = 0; j < 16; ++j) nsq += red[m * 16 + j];

    // squash: v = (nsq/(1+nsq)) * s / sqrt(nsq)
    const float scale = nsq / ((1.0f + nsq) * sqrtf(nsq));
    const float v = s * scale;

    if (tid < 256) {
        const int b = btile * 16 + m;
        out[(size_t)n * BATCH * COUT_ + (size_t)b * COUT_ + (t & 15)] = v;
    }
}

extern "C" void kernel_launch(void* const* d_in, const int* in_sizes, int n_in,
                              void* d_out, int out_size, void* d_ws, size_t ws_size,
                              hipStream_t stream) {
    (void)in_sizes; (void)n_in; (void)d_ws; (void)ws_size; (void)out_size;
    const float* x = (const float*)d_in[0];   // [256, 1152, 8]
    const float* W = (const float*)d_in[1];   // [10, 1152, 8, 16]
    float* out = (float*)d_out;               // [10, 256, 1, 1, 16]

    dim3 grid(NCAPS * (BATCH / 16));          // 160 blocks
    dim3 block(512);                          // 16 wave32
    caps_wmma_kernel<<<grid, block, 0, stream>>>(x, W, out);
}